// MRLRFeatureExtractor_30597347016995
// MI455X (gfx1250) — compile-verified
//
#include <hip/hip_runtime.h>

// MRLR feature extractor for MI455X (gfx1250, wave32).
// One workgroup per batch item. Large GEMMs via V_WMMA_F32_16X16X4_F32; RxR products (A^T A, Newton-Schulz) on VALU.
// B-factor panels staged into LDS with async global->LDS copies (ASYNCcnt),
// double buffered; X*B and B^T*B fused into a single pass over B.

typedef __attribute__((ext_vector_type(2))) float v2f;
typedef __attribute__((ext_vector_type(4))) float v4f;
typedef __attribute__((ext_vector_type(8))) float v8f;

// async global->LDS builtins take per-width int-vector pointers:
//   param1: vector-of-int addrspace(1)*  (global source)
//   param2: vector-of-int addrspace(3)*  (LDS destination)
typedef __attribute__((__vector_size__(16))) int vi4;
typedef __attribute__((__vector_size__(8))) int vi2;
typedef __attribute__((address_space(1))) vi4 gvi4;
typedef __attribute__((address_space(3))) vi4 lvi4;
typedef __attribute__((address_space(1))) vi2 gvi2;
typedef __attribute__((address_space(3))) vi2 lvi2;

#define MM 96
#define NN 9216
#define ITERS 15
#define NS_ITERS 28
#define KC 96  // K-chunk of B staged in LDS (NN % KC == 0)

#if defined(__has_builtin)
#if __has_builtin(__builtin_amdgcn_global_load_async_to_lds_b128) && \
    __has_builtin(__builtin_amdgcn_global_load_async_to_lds_b64)
#define HAVE_ASYNC_LDS 1
#endif
#endif

#ifdef HAVE_ASYNC_LDS
#if __has_builtin(__builtin_amdgcn_s_wait_asynccnt)
#define WAIT_ASYNC(n) __builtin_amdgcn_s_wait_asynccnt(n)
#else
#define WAIT_ASYNC(n) asm volatile("s_wait_asynccnt %0" ::"i"(n) : "memory")
#endif
#else
#define WAIT_ASYNC(n) ((void)0)
#endif

__device__ __forceinline__ v8f wmma4(v2f a, v2f b, v8f c) {
  // D(16x16,f32) = A(16x4,f32) * B(4x16,f32) + C
  return __builtin_amdgcn_wmma_f32_16x16x4_f32(
      false, a, false, b, (short)0, c, false, false);
}

__device__ __forceinline__ float cleanf(float v) {
  // torch.nan_to_num(nan=0, posinf=1e10, neginf=-1e10)
  if (v != v) return 0.0f;
  if (v == __builtin_inff()) return 1e10f;
  if (v == -__builtin_inff()) return -1e10f;
  return v;
}

// -------- async (or fallback) copy of one B chunk (KC x R) into LDS --------
template <int R>
__device__ __forceinline__ void stage_chunk(float* dst, const float* src,
                                            int tid) {
  constexpr int VW = (R == 32) ? 4 : 2;  // floats per copy
  constexpr int NE = KC * R / VW;        // total copies; NE/256 == 3, uniform
  for (int e = tid; e < NE; e += 256) {
#ifdef HAVE_ASYNC_LDS
    if constexpr (VW == 4)
      __builtin_amdgcn_global_load_async_to_lds_b128(
          (gvi4*)(src + e * VW), (lvi4*)(dst + e * VW), 0, 0);
    else
      __builtin_amdgcn_global_load_async_to_lds_b64(
          (gvi2*)(src + e * VW), (lvi2*)(dst + e * VW), 0, 0);
#else
    if constexpr (VW == 4)
      *(v4f*)(dst + e * VW) = *(const v4f*)(src + e * VW);
    else
      *(v2f*)(dst + e * VW) = *(const v2f*)(src + e * VW);
#endif
  }
}

// -------- fused pass over B: XB_s = X @ B (waves 0-5), V_s = B^T B --------
template <int R>
__device__ __forceinline__ void pass_xb_gram(const float* __restrict__ X,
                                             const float* __restrict__ Bg,
                                             float* Bst, float* XB_s,
                                             float* V_s, int tid, int wave,
                                             int lane) {
  constexpr int RT = R / 16;
  constexpr int NCH = NN / KC;
  const int lm = lane & 15, hi = lane >> 4;

  const bool doXB = (wave < 6);
  const int m0 = wave * 16;
  int g = (wave >= 6) ? (wave - 6) : (wave < 2 ? wave + 2 : -1);
  if (g >= RT * RT) g = -1;
  const int ti = (g >= 0) ? g / RT : 0;
  const int tj = (g >= 0) ? g % RT : 0;

  v8f cxb[RT] = {};
  v8f cg = {};

  stage_chunk<R>(Bst, Bg, tid);  // prefetch chunk 0 into buffer 0
  for (int ch = 0; ch < NCH; ch++) {
    float* cur = Bst + (ch & 1) * (KC * R);
    float* nxt = Bst + ((ch + 1) & 1) * (KC * R);
    if (ch + 1 < NCH) {
      stage_chunk<R>(nxt, Bg + (size_t)(ch + 1) * KC * R, tid);
      WAIT_ASYNC(3);  // chunk ch complete; chunk ch+1 (3 copies/thread) in flight
    } else {
      WAIT_ASYNC(0);
    }
    __syncthreads();  // chunk ch visible to all waves
    const int kb = ch * KC;
    if (doXB) {
      for (int k0 = 0; k0 < KC; k0 += 4) {
        const int ka = k0 + 2 * hi;
        v2f a = *(const v2f*)(X + (m0 + lm) * NN + kb + ka);
        for (int j = 0; j < RT; j++) {
          v2f b;
          b.x = cur[ka * R + j * 16 + lm];
          b.y = cur[(ka + 1) * R + j * 16 + lm];
          cxb[j] = wmma4(a, b, cxb[j]);
        }
      }
    }
    if (g >= 0) {
      for (int k0 = 0; k0 < KC; k0 += 4) {
        const int ka = k0 + 2 * hi;
        v2f a, b;
        a.x = cur[ka * R + ti * 16 + lm];
        a.y = cur[(ka + 1) * R + ti * 16 + lm];
        b.x = cur[ka * R + tj * 16 + lm];
        b.y = cur[(ka + 1) * R + tj * 16 + lm];
        cg = wmma4(a, b, cg);
      }
    }
    __syncthreads();  // done reading chunk ch; its buffer may be re-staged
  }
  if (doXB)
    for (int j = 0; j < RT; j++)
      for (int v = 0; v < 8; v++)
        XB_s[(m0 + v + 8 * hi) * R + j * 16 + lm] = cxb[j][v];
  if (g >= 0)
    for (int v = 0; v < 8; v++)
      V_s[(ti * 16 + v + 8 * hi) * R + tj * 16 + lm] = cg[v];
}

// -------- V = A^T A (RxR), K = 96, A in LDS. Tiny: VALU. --------
template <int R>
__device__ __forceinline__ void gram_A(const float* A_s, float* V, int tid) {
  for (int e = tid; e < R * R; e += 256) {
    const int i = e / R, j = e % R;
    float s = 0.0f;
    for (int k = 0; k < MM; k++) s += A_s[k * R + i] * A_s[k * R + j];
    V[e] = s;
  }
}

// -------- pinv of symmetric PSD V via Newton-Schulz; result -> Yout --------
template <int R>
__device__ __forceinline__ void newton_pinv(const float* V, float* Ya,
                                            float* Yb, float* T, float* red,
                                            float* Yout, int tid) {
  if (tid < R) {
    float s = 0.0f;
    for (int j = 0; j < R; j++) s += fabsf(V[tid * R + j]);
    red[tid] = s;
  }
  __syncthreads();
  if (tid == 0) {
    float mx = 1e-30f;
    for (int i = 0; i < R; i++) mx = fmaxf(mx, red[i]);
    red[32] = 1.0f / (mx * mx);  // 1/(||V||_1*||V||_inf), V symmetric
  }
  __syncthreads();
  const float alpha = red[32];
  for (int e = tid; e < R * R; e += 256) Ya[e] = alpha * V[e];  // Y0 = a*V^T
  __syncthreads();
  float* Yc = Ya;
  float* Yn = Yb;
  for (int it = 0; it < NS_ITERS; it++) {
    for (int e = tid; e < R * R; e += 256) {  // T = V*Yc
      const int i = e / R, j = e % R;
      float s = 0.0f;
      for (int k = 0; k < R; k++) s += V[i * R + k] * Yc[k * R + j];
      T[e] = s;
    }
    __syncthreads();
    for (int e = tid; e < R * R; e += 256) {  // Yn = 2*Yc - Yc*T
      const int i = e / R, j = e % R;
      float s = 0.0f;
      for (int k = 0; k < R; k++) s += Yc[i * R + k] * T[k * R + j];
      Yn[e] = 2.0f * Yc[e] - s;
    }
    __syncthreads();
    float* t = Yc; Yc = Yn; Yn = t;
  }
  for (int e = tid; e < R * R; e += 256) Yout[e] = Yc[e];
}

// -------- A_s = clean(XB_s @ Vinv), all operands LDS-resident --------
template <int R>
__device__ __forceinline__ void mulA(const float* XB_s, const float* Vinv,
                                     float* A_s, int wave, int lane) {
  constexpr int RT = R / 16;
  const int lm = lane & 15, hi = lane >> 4;
  if (wave >= 6) return;  // no barriers below
  const int m0 = wave * 16;
  for (int j = 0; j < RT; j++) {
    v8f acc = {};
    for (int k0 = 0; k0 < R; k0 += 4) {
      const int ka = k0 + 2 * hi;
      v2f a = *(const v2f*)(XB_s + (m0 + lm) * R + ka);
      v2f b;
      b.x = Vinv[ka * R + j * 16 + lm];
      b.y = Vinv[(ka + 1) * R + j * 16 + lm];
      acc = wmma4(a, b, acc);
    }
    for (int v = 0; v < 8; v++)
      A_s[(m0 + v + 8 * hi) * R + j * 16 + lm] = cleanf(acc[v]);
  }
}

// -------- B = clean((X^T @ A) @ Vinv) -> global (NN x R) --------
template <int R>
__device__ __forceinline__ void update_B(const float* __restrict__ X,
                                         const float* A_s, const float* Vinv,
                                         float* __restrict__ Bg, float* stage,
                                         int wave, int lane) {
  constexpr int RT = R / 16;
  const int lm = lane & 15, hi = lane >> 4;
  float* st = stage + wave * 512;  // wave-private 16x32 tile
  for (int nt = wave; nt < NN / 16; nt += 8) {
    const int n0 = nt * 16;
    v8f c[RT] = {};
    for (int k0 = 0; k0 < MM; k0 += 4) {
      const int ka = k0 + 2 * hi;
      v2f a;
      a.x = X[ka * NN + n0 + lm];  // (X^T)[n0+lm, ka]
      a.y = X[(ka + 1) * NN + n0 + lm];
      for (int j = 0; j < RT; j++) {
        v2f b;
        b.x = A_s[ka * R + j * 16 + lm];
        b.y = A_s[(ka + 1) * R + j * 16 + lm];
        c[j] = wmma4(a, b, c[j]);
      }
    }
    for (int j = 0; j < RT; j++)
      for (int v = 0; v < 8; v++)
        st[(v + 8 * hi) * 32 + j * 16 + lm] = c[j][v];
    for (int j = 0; j < RT; j++) {
      v8f acc = {};
      for (int k0 = 0; k0 < R; k0 += 4) {
        const int ka = k0 + 2 * hi;
        v2f a2 = *(const v2f*)(st + lm * 32 + ka);
        v2f b2;
        b2.x = Vinv[ka * R + j * 16 + lm];
        b2.y = Vinv[(ka + 1) * R + j * 16 + lm];
        acc = wmma4(a2, b2, acc);
      }
      for (int v = 0; v < 8; v++)
        Bg[(n0 + v + 8 * hi) * R + j * 16 + lm] = cleanf(acc[v]);
    }
  }
}

// -------- approx = A @ B^T and combine --------
// MODE 0: dst = X - approx (residual).  MODE 1: dst = X - Rg + approx (output).
template <int R, int MODE>
__device__ __forceinline__ void reconstruct(const float* __restrict__ X,
                                            const float* __restrict__ Rg,
                                            const float* A_s,
                                            const float* __restrict__ Bg,
                                            float* __restrict__ dst, int wave,
                                            int lane) {
  const int lm = lane & 15, hi = lane >> 4;
  const int PAIRS = (MM / 16) * (NN / 16);
  for (int p = wave; p < PAIRS; p += 8) {
    const int mt = p / (NN / 16), nt = p % (NN / 16);
    const int m0 = mt * 16, n0 = nt * 16;
    v8f c = {};
    for (int k0 = 0; k0 < R; k0 += 4) {
      const int ka = k0 + 2 * hi;
      v2f a = *(const v2f*)(A_s + (m0 + lm) * R + ka);
      v2f b = *(const v2f*)(Bg + (n0 + lm) * R + ka);  // (B^T)[ka, n0+lm]
      c = wmma4(a, b, c);
    }
    for (int v = 0; v < 8; v++) {
      const int idx = (m0 + v + 8 * hi) * NN + n0 + lm;
      if (MODE == 0)
        dst[idx] = X[idx] - c[v];
      else
        dst[idx] = X[idx] - Rg[idx] + c[v];
    }
  }
}

template <int R>
__device__ __forceinline__ void als(const float* X, float* Bg, float* A_s,
                                    float* XB_s, float* V_s, float* Yp_s,
                                    float* Bst, int tid, int wave, int lane) {
  // Newton-Schulz scratch overlaid on the (idle-at-that-point) stage buffer
  float* Ya = Bst;
  float* Yb = Bst + 1024;
  float* Ts = Bst + 2048;
  float* red = Bst + 3072;
  for (int it = 0; it < ITERS; it++) {
    pass_xb_gram<R>(X, Bg, Bst, XB_s, V_s, tid, wave, lane);
    __syncthreads();
    newton_pinv<R>(V_s, Ya, Yb, Ts, red, Yp_s, tid);
    __syncthreads();
    mulA<R>(XB_s, Yp_s, A_s, wave, lane);
    __syncthreads();
    gram_A<R>(A_s, V_s, tid);
    __syncthreads();
    newton_pinv<R>(V_s, Ya, Yb, Ts, red, Yp_s, tid);
    __syncthreads();
    update_B<R>(X, A_s, Yp_s, Bg, Bst, wave, lane);
    __syncthreads();
  }
}

extern "C" __global__ void __launch_bounds__(256)
mrlr_kernel(const float* __restrict__ x, const float* __restrict__ A01,
            const float* __restrict__ B01, const float* __restrict__ A02,
            const float* __restrict__ B02, float* __restrict__ out,
            float* __restrict__ wsB, float* __restrict__ wsR) {
  __shared__ __align__(16) float Bst[2 * KC * 32];  // 24KB: stage / NS scratch
  __shared__ __align__(16) float A_s[MM * 32];      // 12KB factor A
  __shared__ __align__(16) float XB_s[MM * 32];     // 12KB X@B result
  __shared__ __align__(16) float V_s[32 * 32];      // 4KB Gram
  __shared__ __align__(16) float Yp_s[32 * 32];     // 4KB pinv result

  const int tid = threadIdx.x;
  const int wave = tid >> 5;
  const int lane = tid & 31;
  const int item = blockIdx.x;

  const float* X = x + (size_t)item * MM * NN;
  float* Bg = wsB + (size_t)item * NN * 32;
  float* Rg = wsR + (size_t)item * MM * NN;
  float* O = out + (size_t)item * MM * NN;

  // ---- partition 1: rank 32 on X ----
  for (int e = tid; e < NN * 32 / 4; e += 256)
    *(v4f*)(Bg + e * 4) = *(const v4f*)(B01 + e * 4);
  for (int e = tid; e < MM * 32; e += 256) A_s[e] = A01[e];
  __syncthreads();
  als<32>(X, Bg, A_s, XB_s, V_s, Yp_s, Bst, tid, wave, lane);
  reconstruct<32, 0>(X, nullptr, A_s, Bg, Rg, wave, lane);  // Rg = X - approx1
  __syncthreads();

  // ---- partition 2: rank 16 on residual ----
  for (int e = tid; e < NN * 16 / 4; e += 256)
    *(v4f*)(Bg + e * 4) = *(const v4f*)(B02 + e * 4);
  for (int e = tid; e < MM * 16; e += 256) A_s[e] = A02[e];
  __syncthreads();
  als<16>(Rg, Bg, A_s, XB_s, V_s, Yp_s, Bst, tid, wave, lane);
  // out = (X - Rg) + approx2 = approx1 + approx2
  reconstruct<16, 1>(X, Rg, A_s, Bg, O, wave, lane);
}

extern "C" void kernel_launch(void* const* d_in, const int* in_sizes, int n_in,
                              void* d_out, int out_size, void* d_ws,
                              size_t ws_size, hipStream_t stream) {
  const float* x = (const float*)d_in[0];
  const float* A01 = (const float*)d_in[1];
  const float* B01 = (const float*)d_in[2];
  const float* A02 = (const float*)d_in[3];
  const float* B02 = (const float*)d_in[4];
  float* out = (float*)d_out;

  const int batch = in_sizes[0] / (MM * NN);

  // workspace: [B factors: batch*9216*32 f32][residual: batch*96*9216 f32]
  float* wsB = (float*)d_ws;
  float* wsR = wsB + (size_t)batch * NN * 32;

  hipLaunchKernelGGL(mrlr_kernel, dim3(batch), dim3(256), 0, stream, x, A01,
                     B01, A02, B02, out, wsB, wsR);
}